// SinkhornDistance_82686710382677
// MI455X (gfx1250) — compile-verified
//
#include <hip/hip_runtime.h>

// ---------------------------------------------------------------------------
// Sinkhorn distance, fused WMMA implementation for gfx1250 (MI455X).
//
//   B=16, N=1024, D=64, EPS=0.1, MAX_ITER=100, THRESH=0.1
//
// Design: the 1024x1024 cost matrix C is NEVER materialized; every
// (log)sum-exp pass recomputes 16x16 tiles of  x . y^T  with
// V_WMMA_F32_16X16X4_F32 (K=64 swept in 16 steps).  One persistent kernel
// runs the whole coupled Sinkhorn loop using a software global barrier
// (256 blocks, deterministic fixed-order err reduction).
//
// Data layout: softmaxed x/y are stored K-PAIR-major per batch:
//     P[b][kp][n][2],  kp = K/2   (so a WMMA fragment's two K values for a
// lane are one aligned float2 -> single global_load_b64, fully coalesced
// across each 16-lane half).  ~8.6 MB of workspace: lives in L2 (192 MB).
// ---------------------------------------------------------------------------

#define EPS_F      0.1f
#define INV_EPS_F  10.0f
#define MAX_ITER_I 100
#define THRESH_F   0.1f
#define BB 16
#define NN 1024
#define DD 64
#define NTILES (NN / 16)                 // 64 tiles of 16 along each axis
#define BLOCKS_PER_BATCH 16
#define WAVES_PER_BLOCK 4
#define GRID_MAIN (BB * BLOCKS_PER_BATCH)     // 256 blocks in barrier group
#define THREADS_MAIN (32 * WAVES_PER_BLOCK)   // 128 threads = 4 wave32

typedef __attribute__((ext_vector_type(2))) float v2f;
typedef __attribute__((ext_vector_type(8))) float v8f;

// ---------------- workspace layout (float offsets) -------------------------
constexpr int OFF_XT   = 0;                          // [B][32][1024][2] K-pair softmax(x)
constexpr int OFF_YT   = OFF_XT + BB * DD * NN;      // [B][32][1024][2] K-pair softmax(y)
constexpr int OFF_X2   = OFF_YT + BB * DD * NN;      // [B][1024] row sq-norms
constexpr int OFF_Y2   = OFF_X2 + BB * NN;
constexpr int OFF_U    = OFF_Y2 + BB * NN;           // [B][1024]
constexpr int OFF_V    = OFF_U  + BB * NN;
constexpr int OFF_ERR  = OFF_V  + BB * NN;           // [MAX_ITER][256] err partials
constexpr int OFF_COST = OFF_ERR + MAX_ITER_I * GRID_MAIN;  // [256]
constexpr int OFF_BCTR = OFF_COST + GRID_MAIN;       // (int) barrier counters
constexpr int NUM_BARRIERS = 2 * MAX_ITER_I + 1;
constexpr int OFF_END  = OFF_BCTR + NUM_BARRIERS;
constexpr int ZERO_COUNT = OFF_END - OFF_U;          // region re-zeroed per call

// ---------------------------------------------------------------------------
// init: zero u, v, err slots, cost slots, barrier counters (graph-replay safe)
// ---------------------------------------------------------------------------
__global__ void sink_init_kernel(float* ws) {
  for (int i = blockIdx.x * blockDim.x + threadIdx.x; i < ZERO_COUNT;
       i += gridDim.x * blockDim.x)
    ws[OFF_U + i] = 0.0f;
}

// ---------------------------------------------------------------------------
// softmax rows of x and y; store K-pair-major + row squared norms.
// One wave32 per row of 64 elements: lane owns the K-pair (2*lane, 2*lane+1),
// which is exactly one float2 of the packed layout -> single b64 store.
// ---------------------------------------------------------------------------
__global__ void __launch_bounds__(256)
sink_softmax_kernel(const float* __restrict__ X, const float* __restrict__ Y,
                    float* __restrict__ ws) {
  float* XT = ws + OFF_XT;
  float* YT = ws + OFF_YT;
  float* X2 = ws + OFF_X2;
  float* Y2 = ws + OFF_Y2;

  const int gwave = (blockIdx.x * blockDim.x + threadIdx.x) >> 5;
  const int lane  = threadIdx.x & 31;
  const int total = 2 * BB * NN;
  if (gwave >= total) return;

  const bool isY = gwave >= BB * NN;
  const int row  = isY ? gwave - BB * NN : gwave;
  const float* src = (isY ? Y : X) + row * DD;

  float2 vld = ((const float2*)src)[lane];
  float m = fmaxf(vld.x, vld.y);
  #pragma unroll
  for (int s = 16; s >= 1; s >>= 1) m = fmaxf(m, __shfl_xor(m, s, 32));
  float e0 = __expf(vld.x - m), e1 = __expf(vld.y - m);
  float sum = e0 + e1;
  #pragma unroll
  for (int s = 16; s >= 1; s >>= 1) sum += __shfl_xor(sum, s, 32);
  float inv = 1.0f / sum;
  float o0 = e0 * inv, o1 = e1 * inv;

  const int b = row >> 10;
  const int n = row & (NN - 1);
  // K-pair-major: element pair (2*lane, 2*lane+1) of column n -> slot lane*NN+n
  float2* T = (float2*)((isY ? YT : XT) + b * (DD * NN));
  T[lane * NN + n] = make_float2(o0, o1);

  float sq = o0 * o0 + o1 * o1;
  #pragma unroll
  for (int s = 16; s >= 1; s >>= 1) sq += __shfl_xor(sq, s, 32);
  if (lane == 0) (isY ? Y2 : X2)[row] = sq;
}

// ---------------------------------------------------------------------------
// One fused (log)sum-exp phase over a 16-row tile (i0), sweeping all 64
// 16-col tiles.  AT/BT point at this batch's K-pair-major data.
//   C_ij = r2[i] + c2[j] - 2 * (A_i . B_j)      (WMMA f32 16x16x4, K=64)
//   arg  = (radd[i] + cadd[j] - C_ij) / eps
//   radd[i] <- eps*(log_mu - log(sum_j exp(arg) + 1e-6)) + radd[i]
// Fragment addressing: lane half h needs K = 4*k4 + 2*h (+1)  ==  pair index
// kp = 2*k4 + h  ->  one aligned float2 load per fragment.
// Returns |delta u| partial (nonzero only on lanes 0 and 16).
// ---------------------------------------------------------------------------
__device__ __forceinline__ float
lse_phase(const float* __restrict__ AT, const float* __restrict__ BT,
          const float* __restrict__ r2arr, const float* __restrict__ c2arr,
          float* __restrict__ radd, const float* __restrict__ cadd,
          int i0, int lane, float log_mu) {
  const int half = lane >> 4;
  const int l15  = lane & 15;
  const v2f* Ap = (const v2f*)AT;
  const v2f* Bp = (const v2f*)BT;

  v2f afrag[16];
  #pragma unroll
  for (int k4 = 0; k4 < 16; ++k4)
    afrag[k4] = Ap[(k4 * 2 + half) * NN + i0 + l15];

  float r2v[8], rold[8], esum[8];
  #pragma unroll
  for (int r = 0; r < 8; ++r) {
    const int row = i0 + r + 8 * half;
    r2v[r]  = r2arr[row];
    rold[r] = radd[row];
    esum[r] = 0.0f;
  }

  for (int jt = 0; jt < NTILES; ++jt) {
    const int j0 = jt * 16;
    if (jt + 1 < NTILES)
      __builtin_prefetch(&Bp[half * NN + j0 + 16 + l15], 0, 1);
    v8f acc = {0.f, 0.f, 0.f, 0.f, 0.f, 0.f, 0.f, 0.f};
    #pragma unroll
    for (int k4 = 0; k4 < 16; ++k4) {
      const v2f bfrag = Bp[(k4 * 2 + half) * NN + j0 + l15];
      acc = __builtin_amdgcn_wmma_f32_16x16x4_f32(
          false, afrag[k4], false, bfrag, (short)0, acc, false, false);
    }
    const float c2v = c2arr[j0 + l15];
    const float cav = cadd[j0 + l15];
    #pragma unroll
    for (int r = 0; r < 8; ++r) {
      const float Cij = r2v[r] + c2v - 2.0f * acc[r];
      esum[r] += __expf((rold[r] + cav - Cij) * INV_EPS_F);
    }
  }

  // reduce each row over its 16-lane half; lanes (l15==0) own the result
  float delta = 0.0f;
  #pragma unroll
  for (int r = 0; r < 8; ++r) {
    float s = esum[r];
    s += __shfl_xor(s, 8, 32);
    s += __shfl_xor(s, 4, 32);
    s += __shfl_xor(s, 2, 32);
    s += __shfl_xor(s, 1, 32);
    const float rn = EPS_F * (log_mu - __logf(s + 1e-6f)) + rold[r];
    if (l15 == 0) {
      radd[i0 + r + 8 * half] = rn;
      delta += fabsf(rn - rold[r]);
    }
  }
  return delta;
}

// ---------------------------------------------------------------------------
// Final transport-cost partial: sum_{i in tile, all j} exp(M_ij) * C_ij
// ---------------------------------------------------------------------------
__device__ __forceinline__ float
cost_phase(const float* __restrict__ AT, const float* __restrict__ BT,
           const float* __restrict__ r2arr, const float* __restrict__ c2arr,
           const float* __restrict__ radd, const float* __restrict__ cadd,
           int i0, int lane) {
  const int half = lane >> 4;
  const int l15  = lane & 15;
  const v2f* Ap = (const v2f*)AT;
  const v2f* Bp = (const v2f*)BT;

  v2f afrag[16];
  #pragma unroll
  for (int k4 = 0; k4 < 16; ++k4)
    afrag[k4] = Ap[(k4 * 2 + half) * NN + i0 + l15];

  float r2v[8], rv[8];
  #pragma unroll
  for (int r = 0; r < 8; ++r) {
    const int row = i0 + r + 8 * half;
    r2v[r] = r2arr[row];
    rv[r]  = radd[row];
  }
  float csum = 0.0f;
  for (int jt = 0; jt < NTILES; ++jt) {
    const int j0 = jt * 16;
    v8f acc = {0.f, 0.f, 0.f, 0.f, 0.f, 0.f, 0.f, 0.f};
    #pragma unroll
    for (int k4 = 0; k4 < 16; ++k4) {
      const v2f bfrag = Bp[(k4 * 2 + half) * NN + j0 + l15];
      acc = __builtin_amdgcn_wmma_f32_16x16x4_f32(
          false, afrag[k4], false, bfrag, (short)0, acc, false, false);
    }
    const float c2v = c2arr[j0 + l15];
    const float cav = cadd[j0 + l15];
    #pragma unroll
    for (int r = 0; r < 8; ++r) {
      const float Cij = r2v[r] + c2v - 2.0f * acc[r];
      csum += __expf((rv[r] + cav - Cij) * INV_EPS_F) * Cij;
    }
  }
  return csum;
}

// ---------------------------------------------------------------------------
// software global barrier helpers (256 co-resident blocks)
// ---------------------------------------------------------------------------
__device__ __forceinline__ void gbar_arrive_wait(int* ctr) {
  __hip_atomic_fetch_add(ctr, 1, __ATOMIC_RELEASE, __HIP_MEMORY_SCOPE_AGENT);
  while (__hip_atomic_load(ctr, __ATOMIC_ACQUIRE, __HIP_MEMORY_SCOPE_AGENT) <
         GRID_MAIN)
    __builtin_amdgcn_s_sleep(1);
  __threadfence();
}

// ---------------------------------------------------------------------------
// persistent Sinkhorn kernel: the whole while-loop + final cost
// ---------------------------------------------------------------------------
__global__ void __launch_bounds__(THREADS_MAIN)
sink_persistent_kernel(float* __restrict__ ws, float* __restrict__ out) {
  float* XT = ws + OFF_XT;
  float* YT = ws + OFF_YT;
  float* X2 = ws + OFF_X2;
  float* Y2 = ws + OFF_Y2;
  float* U  = ws + OFF_U;
  float* V  = ws + OFF_V;
  float* ERR   = ws + OFF_ERR;
  float* COSTS = ws + OFF_COST;
  int*   BCTR  = (int*)(ws + OFF_BCTR);

  const int b    = blockIdx.x / BLOCKS_PER_BATCH;
  const int blk  = blockIdx.x % BLOCKS_PER_BATCH;
  const int wid  = threadIdx.x >> 5;
  const int lane = threadIdx.x & 31;
  const int tile = blk * WAVES_PER_BLOCK + wid;   // 0..63
  const int i0   = tile * 16;
  const float log_mu = __logf(1.0f / (float)NN + 1e-8f);

  const float* xt = XT + b * (DD * NN);
  const float* yt = YT + b * (DD * NN);
  const float* x2 = X2 + b * NN;
  const float* y2 = Y2 + b * NN;
  float* ub = U + b * NN;
  float* vb = V + b * NN;

  __shared__ float lred[WAVES_PER_BLOCK];
  __shared__ float bcast;

  for (int it = 0; it < MAX_ITER_I; ++it) {
    // ---- phase 1: u update (row lse); accumulate |delta u| ----
    float delta = lse_phase(xt, yt, x2, y2, ub, vb, i0, lane, log_mu);
    #pragma unroll
    for (int s = 16; s >= 1; s >>= 1) delta += __shfl_xor(delta, s, 32);
    if (lane == 0) lred[wid] = delta;
    __threadfence();     // publish u rows before the global barrier
    __syncthreads();
    if (threadIdx.x == 0) {
      float bs = lred[0] + lred[1] + lred[2] + lred[3];
      ERR[it * GRID_MAIN + blockIdx.x] = bs;
      __threadfence();
      gbar_arrive_wait(&BCTR[2 * it]);
    }
    __syncthreads();     // all threads wait for cross-block u visibility

    // ---- phase 2: v update (column lse == row lse of transposed M) ----
    lse_phase(yt, xt, y2, x2, vb, ub, i0, lane, log_mu);
    __threadfence();
    __syncthreads();
    if (threadIdx.x == 0) {
      gbar_arrive_wait(&BCTR[2 * it + 1]);
      float errsum = 0.0f;
      for (int s = 0; s < GRID_MAIN; ++s)      // fixed order: deterministic
        errsum += ERR[it * GRID_MAIN + s];
      bcast = errsum * (1.0f / (float)BB);
    }
    __syncthreads();
    if (bcast < THRESH_F) break;               // same decision in every block
  }

  // ---- final cost: sum exp(M)*C ----
  float csum = cost_phase(xt, yt, x2, y2, ub, vb, i0, lane);
  #pragma unroll
  for (int s = 16; s >= 1; s >>= 1) csum += __shfl_xor(csum, s, 32);
  if (lane == 0) lred[wid] = csum;
  __syncthreads();
  if (threadIdx.x == 0) {
    COSTS[blockIdx.x] = lred[0] + lred[1] + lred[2] + lred[3];
    __threadfence();
    __hip_atomic_fetch_add(&BCTR[2 * MAX_ITER_I], 1, __ATOMIC_RELEASE,
                           __HIP_MEMORY_SCOPE_AGENT);
    if (blockIdx.x == 0) {
      while (__hip_atomic_load(&BCTR[2 * MAX_ITER_I], __ATOMIC_ACQUIRE,
                               __HIP_MEMORY_SCOPE_AGENT) < GRID_MAIN)
        __builtin_amdgcn_s_sleep(1);
      __threadfence();
      float total = 0.0f;
      for (int s = 0; s < GRID_MAIN; ++s) total += COSTS[s];  // fixed order
      out[0] = total * (1.0f / (float)BB);                    // mean over B
    }
  }
}

// ---------------------------------------------------------------------------
extern "C" void kernel_launch(void* const* d_in, const int* in_sizes, int n_in,
                              void* d_out, int out_size, void* d_ws,
                              size_t ws_size, hipStream_t stream) {
  (void)in_sizes; (void)n_in; (void)out_size; (void)ws_size;
  const float* x = (const float*)d_in[0];   // [B,N,D] fp32
  const float* y = (const float*)d_in[1];   // [B,N,D] fp32
  float* out = (float*)d_out;               // scalar fp32
  float* ws  = (float*)d_ws;                // ~8.6 MB used

  sink_init_kernel<<<128, 256, 0, stream>>>(ws);
  // 2*B*N = 32768 rows, one wave32 per row, 8 waves per 256-thread block
  sink_softmax_kernel<<<(2 * BB * NN) / 8, 256, 0, stream>>>(x, y, ws);
  sink_persistent_kernel<<<GRID_MAIN, THREADS_MAIN, 0, stream>>>(ws, out);
}